// PaletteBoundLoss_13434657702363
// MI455X (gfx1250) — compile-verified
//
#include <hip/hip_runtime.h>
#include <math.h>

// PaletteBoundLoss for MI455X (gfx1250, wave32).
// One streaming pass over 1M points. The 14 dot products per point
// (8 face-plane tests + 6 vertex-distance terms) run on the matrix pipe with
// V_WMMA_F32_16X16X4_F32 (A = constant 16x4 hull matrix, B = 16 points as
// columns [x,y,z,1]); the branchy closest-point-on-triangle math runs on the
// VALU, split 4 triangles per half-wave. Accumulation is fully branchless
// (select-weights into FMA/max) to keep the wave32 hot loop free of EXEC-mask
// diamonds. Hierarchical reduction: registers -> LDS -> per-block partials in
// d_ws -> single-block finalize kernel.

typedef __attribute__((ext_vector_type(2))) float v2f;
typedef __attribute__((ext_vector_type(8))) float v8f;

#define W_IN_C  0.001f
#define W_OUT_C 1.0f
#define TOL_C   1e-8f
#define NBLK    1024
#define BLK     256

// Exact port of _closest_point_on_tris (same override order), returning |p-res|^2.
__device__ __forceinline__ float tri_sq(float px, float py, float pz,
                                        float ax, float ay, float az,
                                        float bx, float by, float bz,
                                        float cx, float cy, float cz)
{
    const float eps = 1e-20f;
    float abx = bx - ax, aby = by - ay, abz = bz - az;
    float acx = cx - ax, acy = cy - ay, acz = cz - az;
    float apx = px - ax, apy = py - ay, apz = pz - az;
    float d1 = abx*apx + aby*apy + abz*apz;
    float d2 = acx*apx + acy*apy + acz*apz;
    float bpx = px - bx, bpy = py - by, bpz = pz - bz;
    float d3 = abx*bpx + aby*bpy + abz*bpz;
    float d4 = acx*bpx + acy*bpy + acz*bpz;
    float cpx = px - cx, cpy = py - cy, cpz = pz - cz;
    float d5 = abx*cpx + aby*cpy + abz*cpz;
    float d6 = acx*cpx + acy*cpy + acz*cpz;
    float vc = d1*d4 - d3*d2;
    float vb = d5*d2 - d1*d6;
    float va = d3*d6 - d5*d4;
    float denom = va + vb + vc;
    denom = (fabsf(denom) > eps) ? denom : 1.0f;
    float v = vb / denom;
    float w = vc / denom;
    float rx = ax + abx*v + acx*w;
    float ry = ay + aby*v + acy*w;
    float rz = az + abz*v + acz*w;

    float d43 = d4 - d3, d56 = d5 - d6;
    float sBC = d43 + d56; sBC = (fabsf(sBC) > eps) ? sBC : 1.0f;
    float tBC = d43 / sBC;
    if ((va <= 0.f) && (d43 >= 0.f) && (d56 >= 0.f)) {
        rx = bx + tBC*(cx - bx); ry = by + tBC*(cy - by); rz = bz + tBC*(cz - bz);
    }
    float sAC = d2 - d6; sAC = (fabsf(sAC) > eps) ? sAC : 1.0f;
    float tAC = d2 / sAC;
    if ((vb <= 0.f) && (d2 >= 0.f) && (d6 <= 0.f)) {
        rx = ax + tAC*acx; ry = ay + tAC*acy; rz = az + tAC*acz;
    }
    float sAB = d1 - d3; sAB = (fabsf(sAB) > eps) ? sAB : 1.0f;
    float tAB = d1 / sAB;
    if ((vc <= 0.f) && (d1 >= 0.f) && (d3 <= 0.f)) {
        rx = ax + tAB*abx; ry = ay + tAB*aby; rz = az + tAB*abz;
    }
    if ((d6 >= 0.f) && (d5 <= d6)) { rx = cx; ry = cy; rz = cz; }
    if ((d3 >= 0.f) && (d4 <= d3)) { rx = bx; ry = by; rz = bz; }
    if ((d1 <= 0.f) && (d2 <= 0.f)) { rx = ax; ry = ay; rz = az; }

    float dx = px - rx, dy = py - ry, dz = pz - rz;
    return dx*dx + dy*dy + dz*dz;
}

__global__ __launch_bounds__(BLK)
void pbl_points_kernel(const float* __restrict__ pts,   // N x 3
                       const float* __restrict__ vrt,   // 6 x 3
                       const float* __restrict__ tris,  // 8 x 3 x 3
                       const float* __restrict__ nrm,   // 8 x 3
                       const float* __restrict__ off,   // 8
                       int N, float* __restrict__ ws)
{
    const int tid  = threadIdx.x;
    const int lane = tid & 31;
    const int m    = lane & 15;
    const bool hi  = lane >= 16;

    // ---- Constant A fragment (16x4 f32): lane l holds A[l&15][K], K={0,1}
    // low half / {2,3} high half (per CDNA5 ISA 32-bit A 16x4 layout).
    // Rows 0-7: [nx, ny, nz, -off]  -> D = p.n - off  (inside test)
    // Rows 8-13: [-2vx, -2vy, -2vz, |v|^2] -> D = -2 p.v + |v|^2 (add |p|^2 for d2v)
    float a0 = 0.f, a1 = 0.f;
    if (m < 8) {
        if (!hi) { a0 = nrm[m*3 + 0]; a1 = nrm[m*3 + 1]; }
        else     { a0 = nrm[m*3 + 2]; a1 = -off[m]; }
    } else if (m < 14) {
        int j = m - 8;
        float vx = vrt[j*3 + 0], vy = vrt[j*3 + 1], vz = vrt[j*3 + 2];
        if (!hi) { a0 = -2.f*vx; a1 = -2.f*vy; }
        else     { a0 = -2.f*vz; a1 = vx*vx + vy*vy + vz*vz; }
    }
    v2f afrag; afrag[0] = a0; afrag[1] = a1;

    // ---- Hoist this half-wave's 4 triangles into registers (loop invariant).
    const int tb = hi ? 4 : 0;
    float T[4][9];
#pragma unroll
    for (int t = 0; t < 4; ++t)
#pragma unroll
        for (int q = 0; q < 9; ++q)
            T[t][q] = tris[(tb + t)*9 + q];

    float accS = 0.f;   // sum of d2v over inside points
    float accC = 0.f;   // inside count
    float accM = 0.f;   // max sq over outside points

    const int wavesPerBlock = BLK >> 5;
    const int waveId     = blockIdx.x * wavesPerBlock + (tid >> 5);
    const int totalWaves = gridDim.x * wavesPerBlock;
    const int stride     = totalWaves * 16;

    for (int base = waveId * 16; base < N; base += stride) {
        int pidx = base + m;                       // both halves load same 16 pts
        int pc   = (pidx < N) ? pidx : (N - 1);    // clamp; EXEC stays all-ones
        const float* pp = pts + 3*pc;
        float px = pp[0], py = pp[1], pz = pp[2];
        if (base + stride < N)
            __builtin_prefetch(pts + 3*(pc + stride), 0, 1);  // global_prefetch_b8

        // B fragment (4x16 f32): V0 = {low: K0=x, high: K2=z}, V1 = {low: K1=y, high: K3=1}
        v2f bfrag; bfrag[0] = hi ? pz : px; bfrag[1] = hi ? 1.0f : py;
        v8f cz = {};
        v8f d = __builtin_amdgcn_wmma_f32_16x16x4_f32(
            false, afrag, false, bfrag, (short)0, cz, false, false);

        // Low lanes: d[0..7] = face dots (rows 0-7) for point (base+lane).
        float dmax = fmaxf(fmaxf(fmaxf(d[0], d[1]), fmaxf(d[2], d[3])),
                           fmaxf(fmaxf(d[4], d[5]), fmaxf(d[6], d[7])));
        bool inside = (dmax <= TOL_C);             // meaningful on low lanes

        // High lanes: d[0..5] = rows 8-13 = -2p.v + |v|^2 for the same point.
        float vmin = fminf(fminf(fminf(d[0], d[1]), fminf(d[2], d[3])),
                           fminf(d[4], d[5]));
        float pn2  = px*px + py*py + pz*pz;
        float d2v_hi = pn2 + vmin;                 // meaningful on high lanes

        // Triangles: low half does tris 0-3, high half tris 4-7 of same point.
        float sq = 1e30f;
#pragma unroll
        for (int t = 0; t < 4; ++t)
            sq = fminf(sq, tri_sq(px, py, pz,
                                  T[t][0], T[t][1], T[t][2],
                                  T[t][3], T[t][4], T[t][5],
                                  T[t][6], T[t][7], T[t][8]));

        // Combine halves: low lanes gather d2v and the other 4-tri min.
        float send  = hi ? d2v_hi : 0.0f;
        float d2v   = __shfl_xor(send, 16);
        float sqo   = __shfl_xor(sq, 16);
        float sqAll = fminf(sq, sqo);

        // Branchless accumulation (low lanes only carry valid results).
        bool  active = (!hi) && (pidx < N);
        float wIn    = (active &&  inside) ? 1.0f : 0.0f;
        float wOut   = (active && !inside) ? 1.0f : 0.0f;
        accS += wIn * d2v;
        accC += wIn;
        accM  = fmaxf(accM, wOut * sqAll);
    }

    // ---- Block reduction in LDS, one partial triple per block.
    __shared__ float sS[BLK], sC[BLK], sM[BLK];
    sS[tid] = accS; sC[tid] = accC; sM[tid] = accM;
    __syncthreads();
    for (int o = BLK >> 1; o > 0; o >>= 1) {
        if (tid < o) {
            sS[tid] += sS[tid + o];
            sC[tid] += sC[tid + o];
            sM[tid]  = fmaxf(sM[tid], sM[tid + o]);
        }
        __syncthreads();
    }
    if (tid == 0) {
        ws[3*blockIdx.x + 0] = sS[0];
        ws[3*blockIdx.x + 1] = sC[0];
        ws[3*blockIdx.x + 2] = sM[0];
    }
}

__global__ __launch_bounds__(BLK)
void pbl_finish_kernel(const float* __restrict__ ws, int nblk, int N,
                       float* __restrict__ out)
{
    const int tid = threadIdx.x;
    float s = 0.f, c = 0.f, mx = 0.f;
    for (int i = tid; i < nblk; i += BLK) {
        s += ws[3*i + 0];
        c += ws[3*i + 1];
        mx = fmaxf(mx, ws[3*i + 2]);
    }
    __shared__ float sS[BLK], sC[BLK], sM[BLK];
    sS[tid] = s; sC[tid] = c; sM[tid] = mx;
    __syncthreads();
    for (int o = BLK >> 1; o > 0; o >>= 1) {
        if (tid < o) {
            sS[tid] += sS[tid + o];
            sC[tid] += sC[tid + o];
            sM[tid]  = fmaxf(sM[tid], sM[tid + o]);
        }
        __syncthreads();
    }
    if (tid == 0) {
        float n_in  = sC[0];
        float n_out = (float)N - n_in;
        float loss_in  = (n_in  > 0.f) ? (W_IN_C * sS[0] / fmaxf(n_in, 1.f)) : 0.f;
        float loss_out = (n_out > 0.f) ? (W_OUT_C * sM[0]) : 0.f;
        out[0] = loss_in + loss_out;
    }
}

extern "C" void kernel_launch(void* const* d_in, const int* in_sizes, int n_in,
                              void* d_out, int out_size, void* d_ws, size_t ws_size,
                              hipStream_t stream) {
    const float* pts  = (const float*)d_in[0];  // N x 3
    const float* vrt  = (const float*)d_in[1];  // 6 x 3
    const float* tris = (const float*)d_in[2];  // 8 x 3 x 3
    const float* nrm  = (const float*)d_in[3];  // 8 x 3
    const float* off  = (const float*)d_in[4];  // 8
    const int N = in_sizes[0] / 3;
    float* ws = (float*)d_ws;                   // needs 3*NBLK floats = 12 KB

    pbl_points_kernel<<<dim3(NBLK), dim3(BLK), 0, stream>>>(pts, vrt, tris, nrm, off, N, ws);
    pbl_finish_kernel<<<dim3(1), dim3(BLK), 0, stream>>>(ws, NBLK, N, (float*)d_out);
}